// FlashFFTConvInterface_9131100471888
// MI455X (gfx1250) — compile-verified
//
#include <hip/hip_runtime.h>
#include <hip/hip_bf16.h>

// ============================================================================
// FlashFFTConv for MI455X (gfx1250, wave32, WMMA).
// y[b,c] = circular_conv2d(u[b,c], k[c]) / 512  (H=W=512, ortho rfft2 chain)
// 512-pt FFT = 16x32 Cooley-Tukey, small DFTs executed with
// v_wmma_f32_16x16x4_f32 (the fp32 WMMA on CDNA5). One wave32 per transform.
// Tile traffic uses GLOBAL_LOAD_ASYNC_TO_LDS (ASYNCcnt) when available, with
// the K-spectrum DMA overlapped under the forward-FFT WMMAs in pass 2b.
// ============================================================================

typedef float v2f __attribute__((ext_vector_type(2)));
typedef float v8f __attribute__((ext_vector_type(8)));
typedef int   vi4 __attribute__((vector_size(16)));   // matches builtin's V4i

#if defined(__has_builtin)
#if __has_builtin(__builtin_amdgcn_global_load_async_to_lds_b32) && \
    __has_builtin(__builtin_amdgcn_global_load_async_to_lds_b128) && \
    __has_builtin(__builtin_amdgcn_s_wait_asynccnt)
#define HAVE_ASYNC_LDS 1
#endif
#endif
#ifndef HAVE_ASYNC_LDS
#define HAVE_ASYNC_LDS 0
#endif

#define NW    4            // waves per workgroup (wave32)
#define BLOCK (NW * 32)
#define TSTR  33           // padded stride of 16x32 stage buffer (bank conflicts)
#define TSZ   (16 * TSTR)  // 528 floats

struct SharedFFT {
  float csr[512];          // cos(-2*pi*m/512)
  float csi[512];          // sin(-2*pi*m/512)
  float xr[NW][512];       // per-wave working row (real)
  float xi[NW][512];       // per-wave working row (imag)
  float tr[NW][TSZ];       // per-wave FFT stage scratch (real)
  float ti[NW][TSZ];       // per-wave FFT stage scratch (imag)
  float kr[NW][512];       // K-spectrum tile (real) -- async DMA target
  float ki[NW][512];       // K-spectrum tile (imag)
};

__device__ __forceinline__ void async_wait() {
#if HAVE_ASYNC_LDS
  __builtin_amdgcn_s_wait_asynccnt(0);
#endif
}

// One b32 element: global -> LDS (async DMA when available, else sync copy).
__device__ __forceinline__ void ld_async_b32(float* ldst, const float* gsrc) {
#if HAVE_ASYNC_LDS
  __builtin_amdgcn_global_load_async_to_lds_b32(
      (__attribute__((address_space(1))) int*)gsrc,
      (__attribute__((address_space(3))) int*)ldst, 0, 0);
#else
  *ldst = *gsrc;
#endif
}

// One b128 element: global -> LDS.
__device__ __forceinline__ void ld_async_b128(float4* ldst, const float4* gsrc) {
#if HAVE_ASYNC_LDS
  __builtin_amdgcn_global_load_async_to_lds_b128(
      (__attribute__((address_space(1))) vi4*)gsrc,
      (__attribute__((address_space(3))) vi4*)ldst, 0, 0);
#else
  *ldst = *gsrc;
#endif
}

__device__ __forceinline__ v8f wmma4(v2f a, v2f b, v8f c) {
  // D = A(16x4) * B(4x16) + C, fp32.  (neg_a, A, neg_b, B, c_mod, C, reuse_a, reuse_b)
  return __builtin_amdgcn_wmma_f32_16x16x4_f32(false, a, false, b, (short)0, c,
                                               false, false);
}

__device__ __forceinline__ void init_twiddle(SharedFFT& sh) {
  const float w = -6.28318530717958647692f / 512.0f;
  for (int m = threadIdx.x; m < 512; m += BLOCK) {
    float s, c;
    __sincosf(w * (float)m, &s, &c);
    sh.csr[m] = c;
    sh.csi[m] = s;
  }
}

// ---------------------------------------------------------------------------
// One 512-point complex FFT executed by a single wave32 entirely with WMMA.
//   n = 32*n1 + n2 (n1<16, n2<32), k = k1 + 16*k2:
//   Stage A: Y[k1][n2]  = sum_n1 W16^{n1 k1} x[32 n1 + n2]      (F16 * X)
//   Stage B: Y[k1][n2] *= W512^{k1 n2}
//   Stage C: Z[k1][k2]  = sum_n2 Y[k1][n2] W32^{n2 k2}          (Y * F32)
//   out[k1 + 16 k2] = Z[k1][k2] * oscale
// dir = +1 forward (e^{-i}), -1 inverse (conjugated twiddles).
// Fragment layouts per CDNA5 ISA 7.12.2 (f32: A lanes0-15 K{0,1}, lanes16-31
// K{2,3}; B lane=N with same half-K split; C/D 8 VGPRs M=v(+8), N=lane&15).
// ---------------------------------------------------------------------------
template <bool IMAG_IN>
__device__ __forceinline__ void wave_fft512(float* xr, float* xi,
                                            float* tr, float* ti,
                                            const float* csr, const float* csi,
                                            int lane, float dir, float oscale) {
  const int lh = lane & 15;
  const int hi = lane >> 4;

  // ---------------- Stage A: C[16x32] = F16[16x16] * Xmat[16x32] ----------
  v8f cre0 = {0,0,0,0,0,0,0,0}, cre1 = {0,0,0,0,0,0,0,0};
  v8f cim0 = {0,0,0,0,0,0,0,0}, cim1 = {0,0,0,0,0,0,0,0};
#pragma unroll
  for (int kk = 0; kk < 4; ++kk) {
    const int n1  = 4 * kk + 2 * hi;               // this lane's K pair
    const int ia0 = (lh * n1 * 32) & 511;          // W16^{k1 n1} = W512^{32 k1 n1}
    const int ia1 = (lh * (n1 + 1) * 32) & 511;
    v2f are  = { csr[ia0], csr[ia1] };
    v2f aim  = { dir * csi[ia0], dir * csi[ia1] };
    v2f aimn = { -aim.x, -aim.y };
#pragma unroll
    for (int t = 0; t < 2; ++t) {
      const int col = t * 16 + lh;
      v2f bre = { xr[32 * n1 + col], xr[32 * (n1 + 1) + col] };
      v8f& cr = t ? cre1 : cre0;
      v8f& ci = t ? cim1 : cim0;
      cr = wmma4(are, bre, cr);
      ci = wmma4(aim, bre, ci);
      if (IMAG_IN) {
        v2f bim = { xi[32 * n1 + col], xi[32 * (n1 + 1) + col] };
        cr = wmma4(aimn, bim, cr);
        ci = wmma4(are, bim, ci);
      }
    }
  }

  // ---------------- Stage B: twiddle W512^{k1 n2}, park in tr/ti ----------
#pragma unroll
  for (int t = 0; t < 2; ++t) {
    v8f cr = t ? cre1 : cre0;
    v8f ci = t ? cim1 : cim0;
#pragma unroll
    for (int v = 0; v < 8; ++v) {
      const int k1 = v + 8 * hi;           // D-matrix: M = v (+8 for hi half)
      const int n2 = t * 16 + lh;          // D-matrix: N = lane&15
      const int iw = (k1 * n2) & 511;
      const float wr = csr[iw], wi = dir * csi[iw];
      const float zr = cr[v], zi = ci[v];
      tr[k1 * TSTR + n2] = zr * wr - zi * wi;
      ti[k1 * TSTR + n2] = zr * wi + zi * wr;
    }
  }

  // ---------------- Stage C: Z[16x32] = Y[16x32] * F32[32x32] -------------
  v8f zre0 = {0,0,0,0,0,0,0,0}, zre1 = {0,0,0,0,0,0,0,0};
  v8f zim0 = {0,0,0,0,0,0,0,0}, zim1 = {0,0,0,0,0,0,0,0};
#pragma unroll
  for (int kk = 0; kk < 8; ++kk) {
    const int n2 = 4 * kk + 2 * hi;        // this lane's K pair (A: M=lh=k1)
    v2f are  = { tr[lh * TSTR + n2], tr[lh * TSTR + n2 + 1] };
    v2f aim  = { ti[lh * TSTR + n2], ti[lh * TSTR + n2 + 1] };
    v2f aimn = { -aim.x, -aim.y };
#pragma unroll
    for (int t = 0; t < 2; ++t) {
      const int k2  = t * 16 + lh;
      const int ib0 = (16 * n2 * k2) & 511;        // W32^{n2 k2} = W512^{16 n2 k2}
      const int ib1 = (16 * (n2 + 1) * k2) & 511;
      v2f bre = { csr[ib0], csr[ib1] };
      v2f bim = { dir * csi[ib0], dir * csi[ib1] };
      v8f& zr = t ? zre1 : zre0;
      v8f& zi = t ? zim1 : zim0;
      zr = wmma4(are, bre, zr);
      zi = wmma4(are, bim, zi);
      zr = wmma4(aimn, bim, zr);
      zi = wmma4(aim, bre, zi);
    }
  }

  // ---------------- Output: x[k1 + 16 k2] = Z, scaled ---------------------
#pragma unroll
  for (int t = 0; t < 2; ++t) {
    v8f zr = t ? zre1 : zre0;
    v8f zi = t ? zim1 : zim0;
#pragma unroll
    for (int v = 0; v < 8; ++v) {
      const int k1 = v + 8 * hi;
      const int k2 = t * 16 + lh;
      xr[k1 + 16 * k2] = zr[v] * oscale;
      xi[k1 + 16 * k2] = zi[v] * oscale;
    }
  }
}

// Cooperative 512xNW tile load into per-wave LDS buffers (async when possible).
__device__ __forceinline__ void tile_load(float (*dR)[512], float (*dI)[512],
                                          const float* gR, const float* gI,
                                          size_t base, int c0, int cl, int rb) {
#pragma unroll
  for (int it = 0; it < 16; ++it) {
    const int r = it * 32 + rb;
    const size_t idx = base + (size_t)r * 272 + c0 + cl;
    ld_async_b32(&dR[cl][r], gR + idx);
    ld_async_b32(&dI[cl][r], gI + idx);
  }
}

// ============================================================================
// Pass 1: forward row rFFT (real input).  One wave per row; store bins 0..271
// (padded, only 0..256 are needed) as split re/im, row stride 272.
// ============================================================================
__global__ void __launch_bounds__(BLOCK)
k_row_fft(const float* __restrict__ src, float* __restrict__ dr,
          float* __restrict__ di) {
  __shared__ SharedFFT sh;
  const int tid = threadIdx.x, lane = tid & 31, w = tid >> 5;
  init_twiddle(sh);

  const int plane = blockIdx.y;
  const int row   = blockIdx.x * NW + w;
  const float* p  = src + ((size_t)plane * 512 + row) * 512;
  if (blockIdx.x + 1 < gridDim.x)
    __builtin_prefetch(p + NW * 512, 0, 1);    // global_prefetch_b8, next group

  float4* xv = (float4*)sh.xr[w];
  const float4* pv = (const float4*)p;
#pragma unroll
  for (int j = 0; j < 4; ++j)                  // 2 KB row, async b128 per lane
    ld_async_b128(&xv[lane + 32 * j], &pv[lane + 32 * j]);
  async_wait();
  __syncthreads();                             // twiddle table + row data ready

  wave_fft512<false>(sh.xr[w], sh.xi[w], sh.tr[w], sh.ti[w],
                     sh.csr, sh.csi, lane, 1.0f, 1.0f);

  const size_t ob = ((size_t)plane * 512 + row) * 272;
  for (int c = lane; c < 272; c += 32) {
    dr[ob + c] = sh.xr[w][c];
    di[ob + c] = sh.xi[w][c];
  }
}

// ============================================================================
// Pass 2a: forward column FFT of k-hat, in place (full 2-D spectrum of k).
// Workgroup cooperatively DMA-loads a 512xNW tile; wave w owns column c0+w.
// ============================================================================
__global__ void __launch_bounds__(BLOCK)
k_col_fft(float* __restrict__ gr, float* __restrict__ gi) {
  __shared__ SharedFFT sh;
  const int tid = threadIdx.x, lane = tid & 31, w = tid >> 5;
  init_twiddle(sh);

  const int plane = blockIdx.y;
  const int c0    = blockIdx.x * NW;
  const size_t pb = (size_t)plane * 512 * 272;
  const int cl    = tid & (NW - 1);
  const int rb    = tid >> 2;                     // NW == 4

  tile_load(sh.xr, sh.xi, gr, gi, pb, c0, cl, rb);
  async_wait();
  __syncthreads();

  wave_fft512<true>(sh.xr[w], sh.xi[w], sh.tr[w], sh.ti[w],
                    sh.csr, sh.csi, lane, 1.0f, 1.0f);
  __syncthreads();

#pragma unroll
  for (int it = 0; it < 16; ++it) {
    const int r = it * 32 + rb;
    gr[pb + (size_t)r * 272 + c0 + cl] = sh.xr[cl][r];
    gi[pb + (size_t)r * 272 + c0 + cl] = sh.xi[cl][r];
  }
}

// ============================================================================
// Pass 2b (fused): forward column FFT of u-hat, pointwise * K-spectrum
// (scale 1/512 for the conv), inverse column FFT (scale 1/512), in place.
// The K-tile DMA is issued BEFORE the forward FFT so it streams into LDS
// underneath ~96 WMMAs (ASYNCcnt overlap).
// ============================================================================
__global__ void __launch_bounds__(BLOCK)
k_col_mul_icol(float* __restrict__ ur, float* __restrict__ ui,
               const float* __restrict__ kr, const float* __restrict__ ki) {
  __shared__ SharedFFT sh;
  const int tid = threadIdx.x, lane = tid & 31, w = tid >> 5;
  init_twiddle(sh);

  const int plane = blockIdx.y;                   // b*64 + c
  const int ch    = plane & 63;                   // channel (C = 64)
  const int c0    = blockIdx.x * NW;
  const size_t ub = (size_t)plane * 512 * 272;
  const size_t kb = (size_t)ch * 512 * 272;
  const int cl    = tid & (NW - 1);
  const int rb    = tid >> 2;

  tile_load(sh.xr, sh.xi, ur, ui, ub, c0, cl, rb);   // u tile
  async_wait();
  __syncthreads();

  tile_load(sh.kr, sh.ki, kr, ki, kb, c0, cl, rb);   // K tile DMA (overlapped)

  wave_fft512<true>(sh.xr[w], sh.xi[w], sh.tr[w], sh.ti[w],
                    sh.csr, sh.csi, lane, 1.0f, 1.0f);   // forward columns

  async_wait();                                    // K tile has landed
  __syncthreads();

  const float s = 1.0f / 512.0f;                   // convolution (ortho) scale
  for (int j = lane; j < 512; j += 32) {
    const float ar = sh.xr[w][j], ai = sh.xi[w][j];
    const float br = sh.kr[w][j], bi = sh.ki[w][j];
    sh.xr[w][j] = (ar * br - ai * bi) * s;
    sh.xi[w][j] = (ar * bi + ai * br) * s;
  }

  wave_fft512<true>(sh.xr[w], sh.xi[w], sh.tr[w], sh.ti[w],
                    sh.csr, sh.csi, lane, -1.0f, 1.0f / 512.0f);  // inverse cols
  __syncthreads();

#pragma unroll
  for (int it = 0; it < 16; ++it) {                // store y-hat tile
    const int r = it * 32 + rb;
    ur[ub + (size_t)r * 272 + c0 + cl] = sh.xr[cl][r];
    ui[ub + (size_t)r * 272 + c0 + cl] = sh.xi[cl][r];
  }
}

// ============================================================================
// Pass 3: inverse row FFT.  Rebuild bins 272..511 by Hermitian symmetry of the
// real output rows, inverse transform (scale 1/512), write real part.
// ============================================================================
__global__ void __launch_bounds__(BLOCK)
k_row_ifft(const float* __restrict__ sr, const float* __restrict__ si,
           float* __restrict__ out) {
  __shared__ SharedFFT sh;
  const int tid = threadIdx.x, lane = tid & 31, w = tid >> 5;
  init_twiddle(sh);
  __syncthreads();

  const int plane = blockIdx.y;
  const int row   = blockIdx.x * NW + w;
  const size_t ib = ((size_t)plane * 512 + row) * 272;
  if (blockIdx.x + 1 < gridDim.x)
    __builtin_prefetch(sr + ib + (size_t)NW * 272, 0, 1);

  for (int c = lane; c < 272; c += 32) {
    sh.xr[w][c] = sr[ib + c];
    sh.xi[w][c] = si[ib + c];
  }
  for (int m = 272 + lane; m < 512; m += 32) {   // Hermitian: X[m] = conj(X[512-m])
    sh.xr[w][m] =  sh.xr[w][512 - m];
    sh.xi[w][m] = -sh.xi[w][512 - m];
  }

  wave_fft512<true>(sh.xr[w], sh.xi[w], sh.tr[w], sh.ti[w],
                    sh.csr, sh.csi, lane, -1.0f, 1.0f / 512.0f);

  const size_t ob = ((size_t)plane * 512 + row) * 512;
  float4* ov = (float4*)(out + ob);
  const float4* xv = (const float4*)sh.xr[w];
#pragma unroll
  for (int j = 0; j < 4; ++j) ov[lane + 32 * j] = xv[lane + 32 * j];
}

// ============================================================================
// Host-side launch.  Workspace layout (floats), split re/im, 272-col rows:
//   Ur[256*512*272] | Ui[...] | Kr[64*512*272] | Ki[...]   (~356 MB total)
// ============================================================================
extern "C" void kernel_launch(void* const* d_in, const int* in_sizes, int n_in,
                              void* d_out, int out_size, void* d_ws,
                              size_t ws_size, hipStream_t stream) {
  (void)in_sizes; (void)n_in; (void)out_size; (void)ws_size;
  const float* u = (const float*)d_in[0];   // [4, 64, 512, 512] f32
  const float* k = (const float*)d_in[1];   // [64, 512, 512] f32
  float* out = (float*)d_out;               // [4, 64, 512, 512] f32

  const int PU = 4 * 64;                    // u planes
  const int PK = 64;                        // k planes
  const size_t planeSz = (size_t)512 * 272;

  float* Ur = (float*)d_ws;
  float* Ui = Ur + (size_t)PU * planeSz;
  float* Kr = Ui + (size_t)PU * planeSz;
  float* Ki = Kr + (size_t)PK * planeSz;

  const dim3 blk(BLOCK);
  k_row_fft     <<<dim3(512 / NW, PU), blk, 0, stream>>>(u, Ur, Ui);
  k_row_fft     <<<dim3(512 / NW, PK), blk, 0, stream>>>(k, Kr, Ki);
  k_col_fft     <<<dim3(272 / NW, PK), blk, 0, stream>>>(Kr, Ki);
  k_col_mul_icol<<<dim3(272 / NW, PU), blk, 0, stream>>>(Ur, Ui, Kr, Ki);
  k_row_ifft    <<<dim3(512 / NW, PU), blk, 0, stream>>>(Ur, Ui, out);
}